// HGTClassfication_70849780514981
// MI455X (gfx1250) — compile-verified
//
#include <hip/hip_runtime.h>

// HGT layer for MI455X (gfx1250, wave32).
#define D_   256   // feature dim (== OUT)
#define T_   3     // node types
#define R_   6     // relation types
#define H_   4     // heads
#define DK_  64    // per-head dim
#define CH_  64    // edges per chunk (amortizes 64KB LDS staging of rel matrices)
#define LN_EPS 1e-5f

typedef float v2f __attribute__((ext_vector_type(2)));
typedef float v8f __attribute__((ext_vector_type(8)));

// Order-preserving float->uint encoding for atomicMax-based segment max.
__device__ __forceinline__ unsigned enc_f32(float f) {
  unsigned u = __float_as_uint(f);
  return (u & 0x80000000u) ? ~u : (u | 0x80000000u);
}
__device__ __forceinline__ float dec_f32(unsigned e) {
  return (e & 0x80000000u) ? __uint_as_float(e ^ 0x80000000u)
                           : __uint_as_float(~e);
}

// CDNA5 async global->LDS copy (ASYNCcnt path), 16B per lane per issue.
__device__ __forceinline__ void async_load_b128(const float* gsrc, unsigned ldsoff) {
  unsigned long long ga = (unsigned long long)(uintptr_t)gsrc;
  asm volatile("global_load_async_to_lds_b128 %0, %1, off"
               :: "v"(ldsoff), "v"(ga) : "memory");
}
__device__ __forceinline__ void wait_async0() {
  asm volatile("s_wait_asynccnt 0x0" ::: "memory");
}

// ---------------------------------------------------------------------------
// Per-node-type linear, M-blocked: one wave computes 64 rows x 16 cols with
// V_WMMA_F32_16X16X4_F32. 4 row-tiles x 3 type-accumulators = 12 independent
// WMMA chains; each B (weight) fragment is reused by 4 A fragments.
// ---------------------------------------------------------------------------
__global__ __launch_bounds__(32)
void typed_linear_wmma(const float* __restrict__ X,        // [N,256]
                       const float* __restrict__ W,        // [3,256,256]
                       const float* __restrict__ Bv,       // [3,256]
                       const int*   __restrict__ ntype,    // [N]
                       const float* __restrict__ rowscale, // [N] or nullptr
                       float* __restrict__ Y,              // [N,256]
                       int N)
{
  const int lane = threadIdx.x;
  const int l16  = lane & 15;
  const int hl   = lane >> 4;              // lane half: 0/1
  const int m0   = blockIdx.x * 64;
  const int n0   = blockIdx.y * 16;

  const float2* xr[4];
  float sc[4];
  #pragma unroll
  for (int t4 = 0; t4 < 4; ++t4) {
    int rowA = m0 + t4 * 16 + l16; if (rowA >= N) rowA = N - 1;
    xr[t4] = (const float2*)(X + (size_t)rowA * D_);
    sc[t4] = rowscale ? rowscale[rowA] : 1.0f;
  }

  v8f acc[4][3];
  #pragma unroll
  for (int t4 = 0; t4 < 4; ++t4)
    #pragma unroll
    for (int t = 0; t < 3; ++t) acc[t4][t] = (v8f){0,0,0,0,0,0,0,0};

  const float* W0 = W + n0 + l16;          // +65536 per type

  // A frag (32-bit A 16x4): lanes 0-15 K=k0,k0+1; lanes 16-31 K=k0+2,k0+3.
  // B frag (4x16): comp0 row k0(+2 hi-half), comp1 row k0+1(+2).
  #pragma unroll 2
  for (int k0 = 0; k0 < D_; k0 += 4) {
    const int kb = k0 + 2 * hl;
    v2f b0, b1, b2;
    b0.x = W0[(size_t)kb * D_];              b0.y = W0[(size_t)(kb + 1) * D_];
    b1.x = W0[65536 + (size_t)kb * D_];      b1.y = W0[65536 + (size_t)(kb + 1) * D_];
    b2.x = W0[131072 + (size_t)kb * D_];     b2.y = W0[131072 + (size_t)(kb + 1) * D_];
    #pragma unroll
    for (int t4 = 0; t4 < 4; ++t4) {
      float2 af = xr[t4][(k0 >> 1) + hl];
      v2f a; a.x = af.x * sc[t4]; a.y = af.y * sc[t4];
      acc[t4][0] = __builtin_amdgcn_wmma_f32_16x16x4_f32(false, a, false, b0, (short)0, acc[t4][0], false, false);
      acc[t4][1] = __builtin_amdgcn_wmma_f32_16x16x4_f32(false, a, false, b1, (short)0, acc[t4][1], false, false);
      acc[t4][2] = __builtin_amdgcn_wmma_f32_16x16x4_f32(false, a, false, b2, (short)0, acc[t4][2], false, false);
    }
  }

  // C/D layout: VGPR v, lanes 0-15 -> row v, lanes 16-31 -> row v+8; col = lane%16.
  const int col = n0 + l16;
  #pragma unroll
  for (int t4 = 0; t4 < 4; ++t4) {
    #pragma unroll
    for (int v = 0; v < 8; ++v) {
      const int row = m0 + t4 * 16 + v + 8 * hl;
      if (row < N) {
        const int t = ntype[row];
        const float a0 = acc[t4][0][v], a1 = acc[t4][1][v], a2 = acc[t4][2][v];
        const float sel = (t == 0) ? a0 : (t == 1) ? a1 : a2;
        Y[(size_t)row * D_ + col] = sel + Bv[t * D_ + col];
      }
    }
  }
}

// ---------------------------------------------------------------------------
__global__ void zero_kernel(uint4* __restrict__ p, long n4) {
  long i = (long)blockIdx.x * blockDim.x + threadIdx.x;
  if (i < n4) p[i] = make_uint4(0u, 0u, 0u, 0u);
}

// --------------------------- edge bucketing by relation --------------------
__global__ void count_kernel(const int* __restrict__ etyp, int* __restrict__ rcnt, int E) {
  const int e = blockIdx.x * blockDim.x + threadIdx.x;
  if (e < E) atomicAdd(&rcnt[etyp[e]], 1);
}

__global__ void prefix_kernel(const int* __restrict__ rcnt, int* __restrict__ bucketoff,
                              int* __restrict__ chunkstart, int* __restrict__ rpos) {
  if (threadIdx.x == 0 && blockIdx.x == 0) {
    int off = 0, ch = 0;
    for (int r = 0; r < R_; ++r) {
      bucketoff[r] = off; chunkstart[r] = ch; rpos[r] = off;
      off += rcnt[r];
      ch  += (rcnt[r] + CH_ - 1) / CH_;
    }
    bucketoff[R_] = off; chunkstart[R_] = ch;
  }
}

__global__ void scatter_kernel(const int* __restrict__ etyp, int* __restrict__ rpos,
                               int* __restrict__ eperm, int E) {
  const int e = blockIdx.x * blockDim.x + threadIdx.x;
  if (e < E) {
    const int p = atomicAdd(&rpos[etyp[e]], 1);
    eperm[p] = e;
  }
}

// ---------------------------------------------------------------------------
// Attention logits over a chunk of CH_ same-relation edges. The relation's 4
// head matrices (64KB) are staged into LDS once per chunk with async copies.
// att[e,h] = (k[src,h,:] @ rel_att[r,h]) . q[dst,h,:] * rel_pri[r,h] / 8
// Thread (h,e2) computes one column; wave32 shuffle reduction (each head ==
// two waves) + tiny LDS combine. Running encoded atomicMax + edge counts.
// ---------------------------------------------------------------------------
__global__ __launch_bounds__(256)
void att_chunk(const float* __restrict__ q, const float* __restrict__ k,
               const float* __restrict__ rel_att, const float* __restrict__ rel_pri,
               const int* __restrict__ eperm, const int* __restrict__ esrc,
               const int* __restrict__ edst,
               const int* __restrict__ bucketoff, const int* __restrict__ chunkstart,
               float* __restrict__ att, unsigned* __restrict__ segmax,
               int* __restrict__ cnt)
{
  __shared__ float Amat[H_ * DK_ * DK_];   // 64KB
  __shared__ float partial[8];
  const int tid = threadIdx.x;
  const int b   = blockIdx.x;

  int cs[R_ + 1];
  #pragma unroll
  for (int i = 0; i <= R_; ++i) cs[i] = chunkstart[i];
  int r = -1;
  #pragma unroll
  for (int i = 0; i < R_; ++i) if (b >= cs[i] && b < cs[i + 1]) r = i;
  if (r < 0) return;  // uniform: block beyond total chunks

  {
    const float* src = rel_att + (size_t)r * H_ * DK_ * DK_;
    #pragma unroll
    for (int j = 0; j < 16; ++j) {
      const int idx = (j * 256 + tid) * 4;                  // float index, 16B units
      async_load_b128(src + idx, (unsigned)(uintptr_t)(&Amat[idx]));
    }
    wait_async0();
  }
  __syncthreads();

  const int h = tid >> 6, e2 = tid & 63;
  const int ebase = bucketoff[r] + (b - cs[r]) * CH_;
  const int eend  = bucketoff[r + 1];
  const float* Ah = Amat + h * DK_ * DK_ + e2;

  for (int i = 0; i < CH_; ++i) {
    const int ei = ebase + i;
    if (ei >= eend) break;                // uniform per block
    const int e = eperm[ei];
    const int src = esrc[e], dst = edst[e];
    const float* kk = k + (size_t)src * D_ + h * DK_;
    float key = 0.f;
    #pragma unroll 8
    for (int d = 0; d < DK_; ++d) key = fmaf(kk[d], Ah[d * DK_], key);
    float p = key * q[(size_t)dst * D_ + h * DK_ + e2];
    #pragma unroll
    for (int off = 16; off >= 1; off >>= 1) p += __shfl_xor(p, off, 32);
    if ((tid & 31) == 0) partial[tid >> 5] = p;
    __syncthreads();
    if (tid < H_) {
      const float a = (partial[2 * tid] + partial[2 * tid + 1])
                      * rel_pri[r * H_ + tid] * 0.125f;     // 1/sqrt(64)
      att[e * H_ + tid] = a;
      atomicMax(&segmax[(size_t)(dst * R_ + r) * H_ + tid], enc_f32(a));
    }
    if (tid == 0) atomicAdd(&cnt[dst * R_ + r], 1);
    __syncthreads();
  }
}

// ---------------------------------------------------------------------------
// ex = exp(att - segmax); segden += ex   (in-place att <- ex)
// ---------------------------------------------------------------------------
__global__ void exp_kernel(float* __restrict__ att, const unsigned* __restrict__ segmax,
                           float* __restrict__ segden,
                           const int* __restrict__ edst, const int* __restrict__ etyp,
                           int E)
{
  const int i = blockIdx.x * blockDim.x + threadIdx.x;
  if (i >= E * H_) return;
  const int e = i >> 2, h = i & 3;
  const size_t seg = (size_t)(edst[e] * R_ + etyp[e]) * H_ + h;
  const float m  = dec_f32(segmax[seg]);
  const float ex = expf(att[i] - m);
  att[i] = ex;
  atomicAdd(&segden[seg], ex);
}

// ---------------------------------------------------------------------------
// Message pass over a chunk of same-relation edges; rel_msg staged in LDS.
// msg[dst, h*64+e2] += w[e,h] * (v[src,h,:] @ rel_msg[r,h])[e2]
// ---------------------------------------------------------------------------
__global__ __launch_bounds__(256)
void msg_chunk(const float* __restrict__ v, const float* __restrict__ rel_msg,
               const float* __restrict__ attex, const float* __restrict__ segden,
               const int* __restrict__ eperm, const int* __restrict__ esrc,
               const int* __restrict__ edst,
               const int* __restrict__ bucketoff, const int* __restrict__ chunkstart,
               float* __restrict__ msg)
{
  __shared__ float Mmat[H_ * DK_ * DK_];   // 64KB
  const int tid = threadIdx.x;
  const int b   = blockIdx.x;

  int cs[R_ + 1];
  #pragma unroll
  for (int i = 0; i <= R_; ++i) cs[i] = chunkstart[i];
  int r = -1;
  #pragma unroll
  for (int i = 0; i < R_; ++i) if (b >= cs[i] && b < cs[i + 1]) r = i;
  if (r < 0) return;

  {
    const float* src = rel_msg + (size_t)r * H_ * DK_ * DK_;
    #pragma unroll
    for (int j = 0; j < 16; ++j) {
      const int idx = (j * 256 + tid) * 4;
      async_load_b128(src + idx, (unsigned)(uintptr_t)(&Mmat[idx]));
    }
    wait_async0();
  }
  __syncthreads();

  const int h = tid >> 6, e2 = tid & 63;
  const int ebase = bucketoff[r] + (b - cs[r]) * CH_;
  const int eend  = bucketoff[r + 1];
  const float* Mh = Mmat + h * DK_ * DK_ + e2;

  for (int i = 0; i < CH_; ++i) {
    const int ei = ebase + i;
    if (ei >= eend) break;                // uniform
    const int e = eperm[ei];
    const int src = esrc[e], dst = edst[e];
    const float w = attex[e * H_ + h] / segden[(size_t)(dst * R_ + r) * H_ + h];
    const float* vv = v + (size_t)src * D_ + h * DK_;
    float val = 0.f;
    #pragma unroll 8
    for (int d = 0; d < DK_; ++d) val = fmaf(vv[d], Mh[d * DK_], val);
    atomicAdd(&msg[(size_t)dst * D_ + tid], w * val);
  }
}

// ---------------------------------------------------------------------------
__global__ void nrel_kernel(const int* __restrict__ cnt, float* __restrict__ inv_nrel,
                            int* __restrict__ hasmsg, int N)
{
  const int n = blockIdx.x * blockDim.x + threadIdx.x;
  if (n >= N) return;
  int c = 0;
  #pragma unroll
  for (int r = 0; r < R_; ++r) c += (cnt[n * R_ + r] > 0) ? 1 : 0;
  hasmsg[n]   = (c > 0) ? 1 : 0;
  inv_nrel[n] = 1.0f / (float)((c > 0) ? c : 1);
}

// ---------------------------------------------------------------------------
// Final: out = LN(trans*alpha + x*(1-alpha)) per type, fallback x if no msg.
// One wave per node; wave32 shuffle reductions.
// ---------------------------------------------------------------------------
__global__ __launch_bounds__(256)
void final_kernel(const float* __restrict__ x, const float* __restrict__ trans,
                  const float* __restrict__ skip, const float* __restrict__ ln_g,
                  const float* __restrict__ ln_b, const int* __restrict__ ntype,
                  const int* __restrict__ hasmsg, float* __restrict__ out, int N)
{
  const int gtid = blockIdx.x * blockDim.x + threadIdx.x;
  const int n    = gtid >> 5;
  const int lane = gtid & 31;
  if (n >= N) return;

  const int   t     = ntype[n];
  const float alpha = 1.0f / (1.0f + expf(-skip[t]));
  const int   hm    = hasmsg[n];

  float vals[8];
  float s = 0.f;
  #pragma unroll
  for (int i = 0; i < 8; ++i) {
    const int c = lane + i * 32;
    const float o = trans[(size_t)n * D_ + c] * alpha
                  + x[(size_t)n * D_ + c] * (1.0f - alpha);
    vals[i] = o;
    s += o;
  }
  #pragma unroll
  for (int off = 16; off >= 1; off >>= 1) s += __shfl_xor(s, off, 32);
  const float mu = s * (1.0f / 256.0f);

  float vs = 0.f;
  #pragma unroll
  for (int i = 0; i < 8; ++i) { const float d = vals[i] - mu; vs += d * d; }
  #pragma unroll
  for (int off = 16; off >= 1; off >>= 1) vs += __shfl_xor(vs, off, 32);
  const float inv = rsqrtf(vs * (1.0f / 256.0f) + LN_EPS);

  #pragma unroll
  for (int i = 0; i < 8; ++i) {
    const int c = lane + i * 32;
    const float nv = (vals[i] - mu) * inv * ln_g[t * D_ + c] + ln_b[t * D_ + c];
    out[(size_t)n * D_ + c] = hm ? nv : x[(size_t)n * D_ + c];
  }
}

// ---------------------------------------------------------------------------
extern "C" void kernel_launch(void* const* d_in, const int* in_sizes, int n_in,
                              void* d_out, int out_size, void* d_ws, size_t ws_size,
                              hipStream_t stream)
{
  const float* x       = (const float*)d_in[0];
  const float* Wk      = (const float*)d_in[1];
  const float* bk      = (const float*)d_in[2];
  const float* Wq      = (const float*)d_in[3];
  const float* bq      = (const float*)d_in[4];
  const float* Wv      = (const float*)d_in[5];
  const float* bv      = (const float*)d_in[6];
  const float* Wa      = (const float*)d_in[7];
  const float* ba      = (const float*)d_in[8];
  const float* rel_pri = (const float*)d_in[9];
  const float* rel_att = (const float*)d_in[10];
  const float* rel_msg = (const float*)d_in[11];
  const float* skip    = (const float*)d_in[12];
  const float* ln_g    = (const float*)d_in[13];
  const float* ln_b    = (const float*)d_in[14];
  const int*   ntype   = (const int*)d_in[15];
  const int*   esrc    = (const int*)d_in[16];
  const int*   edst    = (const int*)d_in[17];
  const int*   etyp    = (const int*)d_in[18];

  const int N = in_sizes[0] / D_;
  const int E = in_sizes[16];

  // Workspace layout. [segmax..rcnt] kept contiguous for one zero-fill pass.
  float* ws = (float*)d_ws;
  size_t off = 0;
  float*    qb         = ws + off; off += (size_t)N * D_;
  float*    kb         = ws + off; off += (size_t)N * D_;
  float*    vb         = ws + off; off += (size_t)N * D_;
  float*    att        = ws + off; off += (size_t)E * H_;
  unsigned* segmax     = (unsigned*)(ws + off); off += (size_t)N * R_ * H_;
  float*    segden     = ws + off; off += (size_t)N * R_ * H_;
  int*      cnt        = (int*)(ws + off); off += (size_t)N * R_;
  float*    msg        = ws + off; off += (size_t)N * D_;
  int*      rcnt       = (int*)(ws + off); off += 8;
  int*      bucketoff  = (int*)(ws + off); off += 8;
  int*      chunkstart = (int*)(ws + off); off += 8;
  int*      rpos       = (int*)(ws + off); off += 8;
  float*    inv_nrel   = ws + off; off += (size_t)N;
  int*      hasmsg     = (int*)(ws + off); off += (size_t)N;
  int*      eperm      = (int*)(ws + off); off += (size_t)E;
  float*    trans      = ws + off; off += (size_t)N * D_;

  // 1) zero segmax/segden/cnt/msg/rcnt (encoded -inf == 0; overrun of <=3
  //    words lands in bucketoff which prefix_kernel rewrites).
  {
    const long words  = (long)N * (2 * R_ * H_ + R_ + D_) + 8;
    const long n4     = (words + 3) / 4;
    zero_kernel<<<dim3((unsigned)((n4 + 255) / 256)), 256, 0, stream>>>((uint4*)segmax, n4);
  }

  // 2) bucket edges by relation: count -> prefix -> scatter.
  count_kernel<<<dim3((unsigned)((E + 255) / 256)), 256, 0, stream>>>(etyp, rcnt, E);
  prefix_kernel<<<1, 1, 0, stream>>>(rcnt, bucketoff, chunkstart, rpos);
  scatter_kernel<<<dim3((unsigned)((E + 255) / 256)), 256, 0, stream>>>(etyp, rpos, eperm, E);

  // 3) K/Q/V per-type linears (WMMA f32, 64-row M-blocks).
  dim3 gL((N + 63) / 64, D_ / 16);
  typed_linear_wmma<<<gL, 32, 0, stream>>>(x, Wk, bk, ntype, nullptr, kb, N);
  typed_linear_wmma<<<gL, 32, 0, stream>>>(x, Wq, bq, ntype, nullptr, qb, N);
  typed_linear_wmma<<<gL, 32, 0, stream>>>(x, Wv, bv, ntype, nullptr, vb, N);

  // 4) Edge attention over relation-sorted chunks (async LDS staging).
  const unsigned maxchunks = (unsigned)(E / CH_ + R_);
  att_chunk<<<dim3(maxchunks), 256, 0, stream>>>(qb, kb, rel_att, rel_pri,
                                                 eperm, esrc, edst,
                                                 bucketoff, chunkstart,
                                                 att, segmax, cnt);

  // 5) exp + segment denominators.
  exp_kernel<<<dim3((unsigned)(((long)E * H_ + 255) / 256)), 256, 0, stream>>>(
      att, segmax, segden, edst, etyp, E);

  // 6) Messages over relation-sorted chunks (async LDS staging).
  msg_chunk<<<dim3(maxchunks), 256, 0, stream>>>(vb, rel_msg, att, segden,
                                                 eperm, esrc, edst,
                                                 bucketoff, chunkstart, msg);

  // 7) Cross-relation mean factor + has_msg.
  nrel_kernel<<<dim3((unsigned)((N + 255) / 256)), 256, 0, stream>>>(cnt, inv_nrel, hasmsg, N);

  // 8) Output per-type linear on t = msg * inv_nrel (scale folded into A frag).
  typed_linear_wmma<<<gL, 32, 0, stream>>>(msg, Wa, ba, ntype, inv_nrel, trans, N);

  // 9) Skip-gate + LayerNorm + no-message fallback.
  final_kernel<<<dim3((unsigned)(((long)N * 32 + 255) / 256)), 256, 0, stream>>>(
      x, trans, skip, ln_g, ln_b, ntype, hasmsg, (float*)d_out, N);
}